// NeuralSplineFourierFilterNNX_62019327754415
// MI455X (gfx1250) — compile-verified
//
#include <hip/hip_runtime.h>
#include <math.h>

// ---------------------------------------------------------------------------
// NeuralSplineFourierFilter for MI455X (gfx1250, wave32).
//
// Kernel 1 (1 wave): tiny sin-MLP on scalar `a` using V_WMMA_F32_16X16X4_F32
// for the 32-wide GEMVs, then collapses the degree-3 B-spline into 15
// per-segment cubic polynomials + 15 boundaries (75 floats in d_ws).
//
// Kernel 2 (16.7M points, bandwidth-bound): float4 grid-stride; per point:
// clamp, 14 compares vs uniform boundaries, one LDS B128 coefficient fetch,
// 3 FMAs. 134 MB traffic @ 23.3 TB/s => ~5.8 us floor.
// ---------------------------------------------------------------------------

#define NSEG 15   // spline segments (k = 3..17)

typedef __attribute__((ext_vector_type(2))) float v2f;
typedef __attribute__((ext_vector_type(8))) float v8f;

// y[n] = sum_k x[k] * W[k*NCOLS + n], x striped one element per lane (lane i
// holds x[i]), W row-major [32][NCOLS], result: lane i returns y[i] (i<32,
// cols >= NCOLS are zero-padded).
//
// WMMA mapping per K-chunk of 4: A(16x4) has every row equal to the x chunk
// (so every row of D equals y), B(4x16) is the weight slab.
// 32-bit A layout: lanes 0-15 -> {K0,K1}, lanes 16-31 -> {K2,K3}.
// 32-bit B layout: lane%16 = N; lanes 0-15 -> rows K0,K1; lanes 16-31 -> K2,K3.
// 32-bit D layout: VGPR0 of lane L = D[M=0 or 8][N=L%16]; rows identical here.
template <int NCOLS>
__device__ __forceinline__ float wmma_gemv32(float xlane, const float* __restrict__ W, int lane)
{
    const int  col = lane & 15;
    const bool hi  = lane >= 16;
    v8f acc0 = {0.f, 0.f, 0.f, 0.f, 0.f, 0.f, 0.f, 0.f};
    v8f acc1 = {0.f, 0.f, 0.f, 0.f, 0.f, 0.f, 0.f, 0.f};
#pragma unroll
    for (int kc = 0; kc < 8; ++kc) {
        const int k0 = kc * 4;
        const float x0 = __shfl(xlane, k0 + 0, 32);
        const float x1 = __shfl(xlane, k0 + 1, 32);
        const float x2 = __shfl(xlane, k0 + 2, 32);
        const float x3 = __shfl(xlane, k0 + 3, 32);
        v2f A;
        A[0] = hi ? x2 : x0;
        A[1] = hi ? x3 : x1;
        const int r0 = hi ? (k0 + 2) : k0;   // K rows held by this half-wave
        // N-chunk [0,16)
        v2f B;
        B[0] = (col < NCOLS) ? W[(r0 + 0) * NCOLS + col] : 0.f;
        B[1] = (col < NCOLS) ? W[(r0 + 1) * NCOLS + col] : 0.f;
        acc0 = __builtin_amdgcn_wmma_f32_16x16x4_f32(false, A, false, B,
                                                     (short)0, acc0, false, false);
        // N-chunk [16,32)
        const int c1 = 16 + col;
        B[0] = (c1 < NCOLS) ? W[(r0 + 0) * NCOLS + c1] : 0.f;
        B[1] = (c1 < NCOLS) ? W[(r0 + 1) * NCOLS + c1] : 0.f;
        acc1 = __builtin_amdgcn_wmma_f32_16x16x4_f32(false, A, false, B,
                                                     (short)0, acc1, false, false);
    }
    // All D rows equal y-chunk; lane L<16 wants y[L] (chunk0), L>=16 y[L] (chunk1).
    return hi ? acc1[0] : acc0[0];
}

__global__ __launch_bounds__(32) void spline_setup_kernel(
    const float* __restrict__ a,
    const float* __restrict__ W1, const float* __restrict__ b1,
    const float* __restrict__ W2, const float* __restrict__ b2,
    const float* __restrict__ Ww, const float* __restrict__ bw,
    const float* __restrict__ Wk, const float* __restrict__ bk,
    float* __restrict__ tab)
{
    const int lane = threadIdx.x;

    __shared__ float s_ak[22];     // padded knot vector
    __shared__ float s_c[18];      // control points (0-prepended)
    __shared__ float s_logit[NSEG];

    // layer 1: net = sin(a * W1 + b1), one element per lane
    const float av = a[0];
    float n1 = sinf(av * W1[lane] + b1[lane]);
    // layer 2: net = sin(net @ W2 + b2)   (WMMA GEMV)
    float y2 = wmma_gemv32<32>(n1, W2, lane);
    float n2 = sinf(y2 + b2[lane]);
    // heads
    float yw = wmma_gemv32<17>(n2, Ww, lane);  // control deltas [17]
    float yk = wmma_gemv32<15>(n2, Wk, lane);  // knot logits   [15]

    if (lane < 17) s_c[lane + 1] = yw + bw[lane];
    if (lane == 0) s_c[0] = 0.f;
    if (lane < NSEG) s_logit[lane] = yk + bk[lane];
    __syncthreads();

    if (lane == 0) {
        // softmax + cumsum -> monotone knots; padded knots ak[22]
        float m = -3.0e38f;
        for (int j = 0; j < NSEG; ++j) m = fmaxf(m, s_logit[j]);
        float e[NSEG], sum = 0.f;
        for (int j = 0; j < NSEG; ++j) { e[j] = __expf(s_logit[j] - m); sum += e[j]; }
        const float inv = 1.f / sum;
        s_ak[0] = s_ak[1] = s_ak[2] = s_ak[3] = 0.f;   // 3 pad zeros + k[0]=0
        float cum = 0.f;
        for (int j = 0; j < NSEG; ++j) { cum += e[j]; s_ak[4 + j] = cum * inv; }
        s_ak[19] = s_ak[20] = s_ak[21] = 1.f;          // 3 pad ones (ak[18]==1 from cumsum)
    }
    __syncthreads();

    // Per-segment symbolic de Boor: collapse to monomial cubic per segment.
    if (lane < NSEG) {
        const int k = lane + 3;
        float p[4][4];   // p[j][i]: coeff of x^i of d[j]
#pragma unroll
        for (int j = 0; j < 4; ++j) {
            p[j][0] = s_c[j + k - 3];
            p[j][1] = 0.f; p[j][2] = 0.f; p[j][3] = 0.f;
        }
#pragma unroll
        for (int r = 1; r <= 3; ++r) {
#pragma unroll
            for (int j = 3; j >= 1; --j) {
                if (j < r) continue;
                const float t_lo = s_ak[j + k - 3];
                const float t_hi = s_ak[j + 1 + k - r];
                const float dt   = t_hi - t_lo;
                const float ia   = (dt != 0.f) ? (1.f / dt) : 0.f;
                const float a1   = ia;            // alpha(x) = a0 + a1*x
                const float a0   = -t_lo * ia;
                const float e0 = p[j][0] - p[j-1][0];
                const float e1 = p[j][1] - p[j-1][1];
                const float e2 = p[j][2] - p[j-1][2];
                const float e3 = p[j][3] - p[j-1][3];
                // d[j] = d[j-1] + alpha * (d[j] - d[j-1])
                p[j][0] = p[j-1][0] + a0 * e0;
                p[j][1] = p[j-1][1] + a0 * e1 + a1 * e0;
                p[j][2] = p[j-1][2] + a0 * e2 + a1 * e1;
                p[j][3] = p[j-1][3] + a0 * e3 + a1 * e2;
            }
        }
        tab[lane * 4 + 0] = p[3][0];
        tab[lane * 4 + 1] = p[3][1];
        tab[lane * 4 + 2] = p[3][2];
        tab[lane * 4 + 3] = p[3][3];
        tab[4 * NSEG + lane] = s_ak[4 + lane];   // boundaries kvec[1..15]
    }
}

__device__ __forceinline__ float eval1(float xv, const float* __restrict__ b,
                                       const float4* __restrict__ s_coef)
{
    float xc = fminf(fmaxf(xv * 0.57735026918962576f, 0.f), 0.9999f);
    int s = 0;
#pragma unroll
    for (int j = 0; j < NSEG - 1; ++j)   // b[14]==1 > xc always; 14 compares suffice
        s += (xc >= b[j]) ? 1 : 0;
    const float4 c = s_coef[s];          // ds_load_b128
    return fmaf(fmaf(fmaf(c.w, xc, c.z), xc, c.y), xc, c.x);
}

__global__ __launch_bounds__(256) void spline_eval_kernel(
    const float* __restrict__ x, const float* __restrict__ tab,
    float* __restrict__ out, int n)
{
    __shared__ float4 s_coef[NSEG];
    const int tid = threadIdx.x;
    if (tid < NSEG) s_coef[tid] = ((const float4*)tab)[tid];
    float b[NSEG];
#pragma unroll
    for (int j = 0; j < NSEG; ++j) b[j] = tab[4 * NSEG + j];  // uniform -> scalar regs
    __syncthreads();

    const int gsize = gridDim.x * blockDim.x;
    const int gtid  = blockIdx.x * blockDim.x + tid;
    const int n4    = n >> 2;

    for (int i = gtid; i < n4; i += gsize) {
        const float4 v = ((const float4*)x)[i];
        float4 o;
        o.x = eval1(v.x, b, s_coef);
        o.y = eval1(v.y, b, s_coef);
        o.z = eval1(v.z, b, s_coef);
        o.w = eval1(v.w, b, s_coef);
        ((float4*)out)[i] = o;
    }
    // remainder (n % 4)
    for (int i = (n4 << 2) + gtid; i < n; i += gsize)
        out[i] = eval1(x[i], b, s_coef);
}

extern "C" void kernel_launch(void* const* d_in, const int* in_sizes, int n_in,
                              void* d_out, int out_size, void* d_ws, size_t ws_size,
                              hipStream_t stream)
{
    const float* x  = (const float*)d_in[0];
    const float* a  = (const float*)d_in[1];
    const float* W1 = (const float*)d_in[2];
    const float* b1 = (const float*)d_in[3];
    const float* W2 = (const float*)d_in[4];
    const float* b2 = (const float*)d_in[5];
    const float* Ww = (const float*)d_in[6];
    const float* bw = (const float*)d_in[7];
    const float* Wk = (const float*)d_in[8];
    const float* bk = (const float*)d_in[9];
    float* out = (float*)d_out;
    float* tab = (float*)d_ws;   // 75 floats: 15*4 coeffs + 15 boundaries

    const int n = in_sizes[0];

    spline_setup_kernel<<<1, 32, 0, stream>>>(a, W1, b1, W2, b2, Ww, bw, Wk, bk, tab);

    const int threads = 256;
    long long n4 = (long long)(n >> 2);
    int blocks = (int)((n4 + (long long)threads * 4 - 1) / ((long long)threads * 4));
    if (blocks < 1) blocks = 1;
    spline_eval_kernel<<<blocks, threads, 0, stream>>>(x, tab, out, n);
}